// Rotation_36352603193870
// MI455X (gfx1250) — compile-verified
//
#include <hip/hip_runtime.h>
#include <cmath>

#define CHANNELS 512
#define IN_CH    4096
#define SPA      9                       // 3x3 taps
#define HALF_CH  (IN_CH/2)               // 2048
#define TOTAL_PAIRS (CHANNELS*HALF_CH)   // 1,048,576
#define P16_COUNT   (TOTAL_PAIRS/16)     // 65,536 wave-iterations of 16 pairs
#define NSTD 2097152.0f                  // 512*4096 std samples

typedef __attribute__((ext_vector_type(2))) float v2f;
typedef __attribute__((ext_vector_type(8))) float v8f;

struct WMat { float w[81]; };            // 9x9 grid-sample stencil, passed by value

__device__ __forceinline__ v8f wmma4(v2f a, v2f b, v8f c) {
  // D(16x16,f32) = A(16x4,f32) x B(4x16,f32) + C
  return __builtin_amdgcn_wmma_f32_16x16x4_f32(false, a, false, b, (short)0, c,
                                               false, false);
}

__device__ __forceinline__ v2f mk2(float x, float y) { v2f v; v.x = x; v.y = y; return v; }

// Branchless (raw*cst)^2 partial for one 16-channel group.
// D layout: lane n<16, VGPR i -> cst[ch=n, s=i]; lane 16+n, VGPR i -> cst[ch=n, s=8+i].
// half0 lane owns taps {0,1,4,5,8}; half1 lane owns taps {2,3,6,7} (x4==0 there).
__device__ __forceinline__ float cst_partial(const v8f& d, bool h,
                                             float x0, float x1, float x2,
                                             float x3, float x4) {
  float c8 = __shfl_xor(d[0], 16, 32);  // at half0: cst[n,8]
  float c2 = __shfl_xor(d[2], 16, 32);  // at half1: cst[n,2]
  float c3 = __shfl_xor(d[3], 16, 32);
  float c6 = __shfl_xor(d[6], 16, 32);
  float c7 = __shfl_xor(d[7], 16, 32);
  float cc0 = h ? c2 : d[0];
  float cc1 = h ? c3 : d[1];
  float cc2 = h ? c6 : d[4];
  float cc3 = h ? c7 : d[5];
  float u0 = x0 * cc0, u1 = x1 * cc1, u2 = x2 * cc2, u3 = x3 * cc3, u4 = x4 * c8;
  return u0*u0 + u1*u1 + u2*u2 + u3*u3 + u4*u4;   // u4==0 at half1 (x4 masked)
}

// ---------------------------------------------------------------------------
// Main pass: each wave iteration handles 16 channel pairs (c, c+2048).
// ---------------------------------------------------------------------------
__global__ __launch_bounds__(256) void rot_main(const float* __restrict__ raw,
                                                float* __restrict__ ws,
                                                WMat wm, int nWaves) {
  const int  lane  = threadIdx.x & 31;
  const int  L     = lane & 15;
  const int  half  = lane >> 4;
  const bool h     = (half != 0);
  const float hmsk = h ? 0.0f : 1.0f;   // 1 on half0 lanes, 0 on half1 lanes

  // A operand = stencil W padded to 16x16, K-chunks of 4 (loop invariant).
  // A 16x4 layout: lane m (0-15): v={A[m,0],A[m,1]}; lane 16+m: v={A[m,2],A[m,3]}.
  v2f a0, a1, a2;
  {
    const int t0 = 0 + 2 * half, t1 = 4 + 2 * half, t2 = 8 + 2 * half;
    a0.x = (L < 9 && t0     < 9) ? wm.w[L * 9 + t0]     : 0.0f;
    a0.y = (L < 9 && t0 + 1 < 9) ? wm.w[L * 9 + t0 + 1] : 0.0f;
    a1.x = (L < 9 && t1     < 9) ? wm.w[L * 9 + t1]     : 0.0f;
    a1.y = (L < 9 && t1 + 1 < 9) ? wm.w[L * 9 + t1 + 1] : 0.0f;
    a2.x = (L < 9 && t2     < 9) ? wm.w[L * 9 + t2]     : 0.0f;
    a2.y = 0.0f;  // t2+1 >= 9 always
  }

  const int waveG = blockIdx.x * (blockDim.x >> 5) + (threadIdx.x >> 5);

  float aC = 0.0f;   // sum (raw*cst)^2
  float aO = 0.0f;   // sum over pairs of (a*b)^2  (x2 applied at finalize)
  float aS = 0.0f;   // sum of per-channel std

  for (int p16 = waveG; p16 < P16_COUNT; p16 += nWaves) {
    const int p    = p16 * 16 + L;
    const int nimg = p >> 11;          // / 2048
    const int c    = p & 2047;
    const float* rlo = raw + ((size_t)nimg * IN_CH + c) * SPA;
    const float* rhi = rlo + (size_t)HALF_CH * SPA;

    const int t0 = 2 * half;
    // Unconditional loads (tap 8 masked per half) -> no exec-mask diamonds.
    float ax0 = rlo[t0], ax1 = rlo[t0 + 1], ax2 = rlo[t0 + 4], ax3 = rlo[t0 + 5];
    float ax4 = rlo[8] * hmsk;
    float bx0 = rhi[t0], bx1 = rhi[t0 + 1], bx2 = rhi[t0 + 4], bx3 = rhi[t0 + 5];
    float bx4 = rhi[8] * hmsk;

    // cst^T = W * X^T via 3 chained K=4 WMMAs, per group
    v8f dA = {0.f,0.f,0.f,0.f,0.f,0.f,0.f,0.f};
    dA = wmma4(a0, mk2(ax0, ax1), dA);
    dA = wmma4(a1, mk2(ax2, ax3), dA);
    dA = wmma4(a2, mk2(ax4, 0.f), dA);

    v8f dB = {0.f,0.f,0.f,0.f,0.f,0.f,0.f,0.f};
    dB = wmma4(a0, mk2(bx0, bx1), dB);
    dB = wmma4(a1, mk2(bx2, bx3), dB);
    dB = wmma4(a2, mk2(bx4, 0.f), dB);

    aC += cst_partial(dA, h, ax0, ax1, ax2, ax3, ax4);
    aC += cst_partial(dB, h, bx0, bx1, bx2, bx3, bx4);

    // dis product (per pair, counted once; x2 at finalize). Tap subsets partition 0..8.
    {
      float q0 = ax0 * bx0, q1 = ax1 * bx1, q2 = ax2 * bx2, q3 = ax3 * bx3, q4 = ax4 * bx4;
      aO += q0*q0 + q1*q1 + q2*q2 + q3*q3 + q4*q4;
    }

    // Per-channel std (ddof=1) over 9 taps; combine the two half-wave partials.
    {
      float s1a = ax0 + ax1 + ax2 + ax3 + ax4;
      float s2a = ax0*ax0 + ax1*ax1 + ax2*ax2 + ax3*ax3 + ax4*ax4;
      float s1b = bx0 + bx1 + bx2 + bx3 + bx4;
      float s2b = bx0*bx0 + bx1*bx1 + bx2*bx2 + bx3*bx3 + bx4*bx4;
      s1a += __shfl_xor(s1a, 16, 32);
      s2a += __shfl_xor(s2a, 16, 32);
      s1b += __shfl_xor(s1b, 16, 32);
      s2b += __shfl_xor(s2b, 16, 32);
      float va = fmaxf(s2a - s1a * s1a * (1.0f / 9.0f), 0.0f) * 0.125f;
      float vb = fmaxf(s2b - s1b * s1b * (1.0f / 9.0f), 0.0f) * 0.125f;
      aS += hmsk * (sqrtf(va) + sqrtf(vb));   // count each channel once (half0 lanes)
    }
  }

  // Wave reduction (fixed tree -> deterministic), then block reduction in LDS.
  for (int off = 16; off > 0; off >>= 1) {
    aC += __shfl_xor(aC, off, 32);
    aO += __shfl_xor(aO, off, 32);
    aS += __shfl_xor(aS, off, 32);
  }
  __shared__ float red[8][3];
  const int wv = threadIdx.x >> 5;
  if (lane == 0) { red[wv][0] = aC; red[wv][1] = aO; red[wv][2] = aS; }
  __syncthreads();
  if (threadIdx.x == 0) {
    float tc = 0.f, to = 0.f, ts = 0.f;
    for (int w = 0; w < 8; ++w) { tc += red[w][0]; to += red[w][1]; ts += red[w][2]; }
    ws[blockIdx.x * 3 + 0] = tc;
    ws[blockIdx.x * 3 + 1] = to;
    ws[blockIdx.x * 3 + 2] = ts;
  }
}

// ---------------------------------------------------------------------------
// Deterministic finalize: sum block partials in a fixed order, emit scalar.
// ---------------------------------------------------------------------------
__global__ __launch_bounds__(256) void rot_final(const float* __restrict__ ws,
                                                 int nBlocks,
                                                 float* __restrict__ out) {
  __shared__ float rc[256], ro[256], rs[256];
  const int tid = threadIdx.x;
  float c = 0.f, o = 0.f, s = 0.f;
  for (int i = tid; i < nBlocks; i += 256) {
    c += ws[3 * i + 0]; o += ws[3 * i + 1]; s += ws[3 * i + 2];
  }
  rc[tid] = c; ro[tid] = o; rs[tid] = s;
  __syncthreads();
  for (int st = 128; st > 0; st >>= 1) {
    if (tid < st) { rc[tid] += rc[tid + st]; ro[tid] += ro[tid + st]; rs[tid] += rs[tid + st]; }
    __syncthreads();
  }
  if (tid == 0) {
    float resCST  = sqrtf(rc[0]);
    float resOth  = sqrtf(2.0f * ro[0]);   // pair symmetry factor
    float losOth  = resOth / (resCST + resOth + 1e-6f);
    float meanStd = rs[0] / NSTD;
    float losSTD  = fmaxf(-meanStd, -0.5f) * 8.0f + 4.0f;
    out[0] = losOth + losSTD;
  }
}

// ---------------------------------------------------------------------------
// Host: build the constant 9x9 grid-sample stencil (theta = pi*511/512),
// matching the reference's float64 grid -> float32 bilinear math.
// ---------------------------------------------------------------------------
static void build_w(float* w) {
  const double theta = M_PI * (CHANNELS - 1) / (double)CHANNELS;
  const double ct = cos(theta), st = sin(theta);
  const double lin[3] = {-1.0, 0.0, 1.0};
  for (int i = 0; i < 81; ++i) w[i] = 0.0f;
  for (int i = 0; i < 3; ++i) {
    for (int j = 0; j < 3; ++j) {
      float gx = (float)(ct * lin[j] - st * lin[i]);
      float gy = (float)(st * lin[j] + ct * lin[i]);
      float ix = gx + 1.0f;   // (gx+1)*(W-1)/2 with W=3
      float iy = gy + 1.0f;
      float x0 = floorf(ix), y0 = floorf(iy);
      float x1 = x0 + 1.0f,  y1 = y0 + 1.0f;
      float w00 = (x1 - ix) * (y1 - iy);
      float w01 = (ix - x0) * (y1 - iy);
      float w10 = (x1 - ix) * (iy - y0);
      float w11 = (ix - x0) * (iy - y0);
      const float ys_[4] = {y0, y0, y1, y1};
      const float xs_[4] = {x0, x1, x0, x1};
      const float ww_[4] = {w00, w01, w10, w11};
      for (int t = 0; t < 4; ++t) {
        bool valid = xs_[t] >= 0.0f && xs_[t] <= 2.0f && ys_[t] >= 0.0f && ys_[t] <= 2.0f;
        int xc = (int)fminf(fmaxf(xs_[t], 0.0f), 2.0f);
        int yc = (int)fminf(fmaxf(ys_[t], 0.0f), 2.0f);
        if (valid) w[(i * 3 + j) * 9 + (yc * 3 + xc)] += ww_[t];
      }
    }
  }
}

extern "C" void kernel_launch(void* const* d_in, const int* in_sizes, int n_in,
                              void* d_out, int out_size, void* d_ws, size_t ws_size,
                              hipStream_t stream) {
  (void)in_sizes; (void)n_in; (void)out_size;
  const float* raw = (const float*)d_in[0];
  float* out = (float*)d_out;
  float* ws  = (float*)d_ws;

  WMat wm;
  build_w(wm.w);

  int nBlocks = 512;
  const size_t need = (size_t)nBlocks * 3 * sizeof(float);
  if (ws_size < need) {
    int nb = (int)(ws_size / (3 * sizeof(float)));
    nBlocks = nb < 1 ? 1 : nb;
  }
  const int nWaves = nBlocks * (256 / 32);

  rot_main<<<nBlocks, 256, 0, stream>>>(raw, ws, wm, nWaves);
  rot_final<<<1, 256, 0, stream>>>(ws, nBlocks, out);
}